// GNNRecommender_6975026889203
// MI455X (gfx1250) — compile-verified
//
#include <hip/hip_runtime.h>
#include <hip/hip_bf16.h>

typedef __attribute__((ext_vector_type(2))) float v2f;
typedef __attribute__((ext_vector_type(8))) float v8f;

#define D 128   // feature dim (both layers)

// ---------------- degree / dinv ----------------
__global__ void init_deg_kernel(float* deg, int n) {
    int i = blockIdx.x * blockDim.x + threadIdx.x;
    if (i < n) deg[i] = 1.0f;                      // self-loop
}

__global__ void count_deg_kernel(float* deg, const int* __restrict__ dst, int nE) {
    int i = blockIdx.x * blockDim.x + threadIdx.x;
    if (i < nE) unsafeAtomicAdd(&deg[dst[i]], 1.0f);
}

__global__ void deg_to_dinv_kernel(float* deg, int n) {
    int i = blockIdx.x * blockDim.x + threadIdx.x;
    if (i < n) {
        float d = deg[i];
        deg[i] = (d > 0.0f) ? rsqrtf(d) : 0.0f;
    }
}

// ---------------- GEMM (fp32 WMMA) + row scale by dinv ----------------
// G[i,:] = (X @ W)[i,:] * dinv[i].  One wave computes a 16-row strip x 128 cols.
// W staged in LDS as float2 pairs: Wlds[k2*128+n] = (W[2k2][n], W[2k2+1][n]),
// so a B fragment (4x16 f32, lanes 0-15: K=kk..kk+1, lanes 16-31: K=kk+2..kk+3)
// is a single ds_load_b64.
__global__ void __launch_bounds__(256) gemm_scale_kernel(
    const float* __restrict__ X, const float* __restrict__ W,
    const float* __restrict__ dinv, float* __restrict__ G, int nRows)
{
    __shared__ v2f Wlds[64 * 128];                 // 64 KB
    const int tid = threadIdx.x;

    // cooperative, coalesced load of W into paired LDS layout
    #pragma unroll
    for (int it = 0; it < 32; ++it) {
        int p  = it * 256 + tid;                   // pair index 0..8191
        int k2 = p >> 7;
        int n  = p & 127;
        v2f w;
        w.x = W[(2 * k2)     * D + n];
        w.y = W[(2 * k2 + 1) * D + n];
        Wlds[p] = w;
    }
    __syncthreads();

    const int wave  = tid >> 5;
    const int lane  = tid & 31;
    const int strip = blockIdx.x * 8 + wave;       // 16-row strip index
    const int nStrips = (nRows + 15) >> 4;
    if (strip >= nStrips) return;

    const int mlane = lane & 15;                   // row within strip (A) / col (B)
    const int khalf = lane >> 4;                   // 0: K 0..1, 1: K 2..3

    const float* xrow = X + (size_t)(strip * 16 + mlane) * D + 2 * khalf;

    v8f acc[8] = {};                               // 8 tiles of 16x16 f32 accum

    for (int kk = 0; kk < D; kk += 4) {
        v2f a = *(const v2f*)(xrow + kk);          // A frag: global_load_b64
        const v2f* brow = &Wlds[((kk >> 1) + khalf) * 128 + mlane];
        #pragma unroll
        for (int nt = 0; nt < 8; ++nt) {
            v2f b = brow[nt * 16];                 // B frag: ds_load_b64
            acc[nt] = __builtin_amdgcn_wmma_f32_16x16x4_f32(
                false, a, false, b, (short)0, acc[nt], false, false);
        }
    }

    // epilogue: D layout -> VGPR r holds row (khalf*8 + r), col = nt*16 + mlane
    const int mbase = strip * 16 + khalf * 8;
    #pragma unroll
    for (int r = 0; r < 8; ++r) {
        float dr = dinv[mbase + r];
        float* grow = G + (size_t)(mbase + r) * D + mlane;
        #pragma unroll
        for (int nt = 0; nt < 8; ++nt)
            grow[nt * 16] = acc[nt][r] * dr;
    }
}

// ---------------- copy g -> acc (self-loop seed), float4 ----------------
__global__ void copy4_kernel(const float* __restrict__ src, float* __restrict__ dstp, int n4) {
    int i = blockIdx.x * blockDim.x + threadIdx.x;
    if (i < n4) ((float4*)dstp)[i] = ((const float4*)src)[i];
}

// ---------------- edge scatter: acc[dst] += g[src], wave per edge ----------------
__global__ void scatter_kernel(const float* __restrict__ g, float* __restrict__ acc,
                               const int* __restrict__ src, const int* __restrict__ dst,
                               int nE)
{
    int t    = blockIdx.x * blockDim.x + threadIdx.x;
    int edge = t >> 5;
    int lane = t & 31;
    if (edge >= nE) return;
    int s = src[edge];
    int d = dst[edge];
    float4 v = ((const float4*)(g + (size_t)s * D))[lane];   // 32 lanes x float4 = 128
    float* ap = acc + (size_t)d * D + lane * 4;
    unsafeAtomicAdd(ap + 0, v.x);
    unsafeAtomicAdd(ap + 1, v.y);
    unsafeAtomicAdd(ap + 2, v.z);
    unsafeAtomicAdd(ap + 3, v.w);
}

// ---------------- combine: acc = relu(dinv[row]*acc + bias[col]), in place ----------------
__global__ void combine_kernel(float* __restrict__ acc, const float* __restrict__ dinv,
                               const float* __restrict__ bias, int n4)
{
    int i = blockIdx.x * blockDim.x + threadIdx.x;   // one float4 per thread, 32 per row
    if (i >= n4) return;
    int row = i >> 5;
    int c4  = i & 31;
    float dr = dinv[row];
    float4 v = ((float4*)acc)[i];
    float4 b = ((const float4*)bias)[c4];
    v.x = fmaxf(fmaf(dr, v.x, b.x), 0.0f);
    v.y = fmaxf(fmaf(dr, v.y, b.y), 0.0f);
    v.z = fmaxf(fmaf(dr, v.z, b.z), 0.0f);
    v.w = fmaxf(fmaf(dr, v.w, b.w), 0.0f);
    ((float4*)acc)[i] = v;
}

// ---------------- FC head: out[i] = h[i,:] . fcW + fcb, wave per node ----------------
__global__ void fc_kernel(const float* __restrict__ h, const float* __restrict__ fcW,
                          const float* __restrict__ fcb, float* __restrict__ out, int n)
{
    int t    = blockIdx.x * blockDim.x + threadIdx.x;
    int row  = t >> 5;
    int lane = t & 31;
    if (row >= n) return;
    float4 hv = ((const float4*)(h + (size_t)row * D))[lane];
    float4 wv = ((const float4*)fcW)[lane];
    float s = hv.x * wv.x + hv.y * wv.y + hv.z * wv.z + hv.w * wv.w;
    #pragma unroll
    for (int off = 16; off >= 1; off >>= 1)
        s += __shfl_xor(s, off, 32);
    if (lane == 0) out[row] = s + fcb[0];
}

extern "C" void kernel_launch(void* const* d_in, const int* in_sizes, int n_in,
                              void* d_out, int out_size, void* d_ws, size_t ws_size,
                              hipStream_t stream) {
    const float* x   = (const float*)d_in[0];
    const int*   ei  = (const int*)d_in[1];
    const float* W1  = (const float*)d_in[2];
    const float* b1  = (const float*)d_in[3];
    const float* W2  = (const float*)d_in[4];
    const float* b2  = (const float*)d_in[5];
    const float* fcW = (const float*)d_in[6];
    const float* fcb = (const float*)d_in[7];

    const int nNodes = in_sizes[0] / D;
    const int nE     = in_sizes[1] / 2;
    const int* src = ei;
    const int* dst = ei + nE;

    float* ws   = (float*)d_ws;
    size_t off  = ((size_t)nNodes + 3) & ~(size_t)3;   // keep 16B alignment
    float* dinv = ws;
    float* bufA = ws + off;                            // g
    float* bufB = bufA + (size_t)nNodes * D;           // acc / h

    const int T = 256;
    const int n4      = nNodes * (D / 4);
    const int gN      = (nNodes + T - 1) / T;
    const int gE      = (nE + T - 1) / T;
    const int g4      = (n4 + T - 1) / T;
    const int gEdgeW  = (nE * 32 + T - 1) / T;         // wave per edge
    const int nStrips = (nNodes + 15) / 16;
    const int gGemm   = (nStrips + 7) / 8;             // 8 waves per block
    const int gFc     = (nNodes * 32 + T - 1) / T;

    // degrees -> dinv (in place)
    init_deg_kernel<<<gN, T, 0, stream>>>(dinv, nNodes);
    count_deg_kernel<<<gE, T, 0, stream>>>(dinv, dst, nE);
    deg_to_dinv_kernel<<<gN, T, 0, stream>>>(dinv, nNodes);

    // ---- layer 1 ----
    gemm_scale_kernel<<<gGemm, T, 0, stream>>>(x, W1, dinv, bufA, nNodes);
    copy4_kernel<<<g4, T, 0, stream>>>(bufA, bufB, n4);
    scatter_kernel<<<gEdgeW, T, 0, stream>>>(bufA, bufB, src, dst, nE);
    combine_kernel<<<g4, T, 0, stream>>>(bufB, dinv, b1, n4);

    // ---- layer 2 (reads bufB=h1, reuses bufA for g2, then bufB becomes h2) ----
    gemm_scale_kernel<<<gGemm, T, 0, stream>>>(bufB, W2, dinv, bufA, nNodes);
    copy4_kernel<<<g4, T, 0, stream>>>(bufA, bufB, n4);
    scatter_kernel<<<gEdgeW, T, 0, stream>>>(bufA, bufB, src, dst, nE);
    combine_kernel<<<g4, T, 0, stream>>>(bufB, dinv, b2, n4);

    // ---- FC head ----
    fc_kernel<<<gFc, T, 0, stream>>>(bufB, fcW, fcb, (float*)d_out, nNodes);
}